// PatchQuantumGenerator_78804059947104
// MI455X (gfx1250) — compile-verified
//
#include <hip/hip_runtime.h>

// ---------------------------------------------------------------------------
// PatchQuantumGenerator on gfx1250:
//   - 16x16 group unitaries applied with V_WMMA_F32_16X16X4_F32 (full f32)
//   - CNOT chain  == Gray-code permutation  new[j] = old[j ^ (j>>1)]
//   - initial RY layer on |0> == rank-1 tensor product, computed directly
//   - statevector lives in LDS (16KB per wave, ping-pong buffers)
// ---------------------------------------------------------------------------

#define NQ        12
#define NSTATE    4096          // 1 << NQ
#define QDEPTH    6
#define NGEN      8
#define KOUT      1024          // patch_size == 1 << (NQ - 2)
#define WPB       2             // waves per block
#define BLOCK     (WPB * 32)

typedef float v2f __attribute__((ext_vector_type(2)));
typedef float v8f __attribute__((ext_vector_type(8)));

#if defined(__HIP_DEVICE_COMPILE__) && defined(__gfx1250__)
#if !__has_builtin(__builtin_amdgcn_wmma_f32_16x16x4_f32)
#error "wmma f32 16x16x4 builtin not available on this toolchain"
#endif
#endif

// ---------------------------------------------------------------------------
// Precompute the 144 fixed 16x16 group unitaries:
//   U[g][l][grp][r][c] = prod_{p=0..3} RY(w[l][11-4*grp-p])[bit_p(r)][bit_p(c)]
// where grp selects flat-index bits [4*grp+3 : 4*grp], qubit q <-> bit (11-q),
// RY(theta) = [[cos(t/2), -sin(t/2)], [sin(t/2), cos(t/2)]].
// ---------------------------------------------------------------------------
__global__ void qgen_precompute_unitaries(const float* __restrict__ qp,
                                          float* __restrict__ U) {
  const int bx  = blockIdx.x;           // ((g*QDEPTH)+l)*3 + grp, 0..143
  const int g   = bx / (QDEPTH * 3);
  const int rem = bx % (QDEPTH * 3);
  const int l   = rem / 3;
  const int grp = rem % 3;
  const int sh  = grp * 4;

  const int tid = threadIdx.x;          // 0..255
  const int r = tid >> 4;
  const int c = tid & 15;

  float val = 1.0f;
#pragma unroll
  for (int p = 0; p < 4; ++p) {
    const int q = 11 - (sh + p);        // qubit owning bit p of this nibble
    const float ang = 0.5f * qp[g * (QDEPTH * NQ) + l * NQ + q];
    const float cs = cosf(ang);
    const float sn = sinf(ang);
    const int br = (r >> p) & 1;
    const int bc = (c >> p) & 1;
    // RY element [br][bc]: [[c, -s], [s, c]]
    const float e = br ? (bc ? cs : sn) : (bc ? -sn : cs);
    val *= e;
  }
  U[bx * 256 + tid] = val;
}

// ---------------------------------------------------------------------------
// Main simulator: one wave32 per (batch element, generator).
// ---------------------------------------------------------------------------
__global__ __launch_bounds__(BLOCK) void qgen_simulate(
    const float* __restrict__ x,       // (B, 12) noise angles
    const float* __restrict__ Umats,   // (8,6,3,16,16) precomputed unitaries
    float* __restrict__ out) {         // (B, 8*1024)
  __shared__ float lds[2][WPB][NSTATE];

  const int lane = threadIdx.x & 31;
  const int w    = threadIdx.x >> 5;
  const int b    = blockIdx.x * WPB + w;
  const int g    = blockIdx.y;

  // ---- per-qubit cos/sin of the data-dependent RY layer --------------------
  float cq[NQ], sq[NQ];
#pragma unroll
  for (int q = 0; q < NQ; ++q) {
    const float ang = 0.5f * x[b * NQ + q];
    cq[q] = cosf(ang);
    sq[q] = sinf(ang);
  }

  // ---- initial state: tensor product of [cos, sin] per qubit ---------------
  // flat index i = t*32 + lane; qubit q <-> bit (11-q) of i.
  float Qlane = 1.0f;
#pragma unroll
  for (int q = 7; q < 12; ++q) {               // bits 4..0 come from lane
    const int bit = (lane >> (11 - q)) & 1;
    Qlane *= bit ? sq[q] : cq[q];
  }
  int cur = 0;
  for (int t = 0; t < 128; ++t) {
    float v = Qlane;
#pragma unroll
    for (int q = 0; q < 7; ++q) {              // bits 11..5 come from t
      const int bit = (t >> (6 - q)) & 1;
      v *= bit ? sq[q] : cq[q];
    }
    lds[cur][w][t * 32 + lane] = v;
  }

  // ---- WMMA fragment lane geometry (ISA 7.12.2, 32-bit A/B/C layouts) ------
  const int m_a   = lane & 15;                 // A: lane 0-15 -> row M
  const int khalf = (lane >> 4) << 1;          // VGPR pair holds K {0,1} or {2,3}
  const int n_b   = lane & 15;                 // B/D: lane 0-15 -> column N
  const int m_off = (lane >> 4) << 3;          // D rows: 0..7 or 8..15

  for (int l = 0; l < QDEPTH; ++l) {
    // -- three 16x16 group unitaries (all RYs in a layer commute) -----------
    for (int grp = 0; grp < 3; ++grp) {
      const int sh = grp * 4;
      const float* __restrict__ U =
          Umats + (((g * QDEPTH) + l) * 3 + grp) * 256;

      // A fragments: 4 chunks of 16x4 from the unitary (row-major)
      v2f afrag[4];
#pragma unroll
      for (int kc = 0; kc < 4; ++kc) {
        const int k0 = kc * 4 + khalf;
        afrag[kc].x = U[m_a * 16 + k0];
        afrag[kc].y = U[m_a * 16 + k0 + 1];
      }

      // 256 contexts per group -> 16 tiles of 16 columns
      for (int t = 0; t < 16; ++t) {
        const int h = t * 16 + n_b;            // context id 0..255
        int base;
        if (grp == 0)      base = h << 4;                       // bits 3:0
        else if (grp == 1) base = ((h >> 4) << 8) | (h & 15);   // bits 7:4
        else               base = h;                            // bits 11:8

        v8f acc = {0.0f, 0.0f, 0.0f, 0.0f, 0.0f, 0.0f, 0.0f, 0.0f};
#pragma unroll
        for (int kc = 0; kc < 4; ++kc) {
          const int k0 = kc * 4 + khalf;
          v2f bfrag;
          bfrag.x = lds[cur][w][base + (k0 << sh)];
          bfrag.y = lds[cur][w][base + ((k0 + 1) << sh)];
          acc = __builtin_amdgcn_wmma_f32_16x16x4_f32(
              /*neg_a=*/false, afrag[kc],
              /*neg_b=*/false, bfrag,
              /*c_mod=*/(short)0, acc,
              /*reuse_a=*/false, /*reuse_b=*/false);
        }
        // scatter D back in place (same-wave LDS ops stay in order)
#pragma unroll
        for (int r = 0; r < 8; ++r) {
          lds[cur][w][base + ((r + m_off) << sh)] = acc[r];
        }
      }
    }

    // -- CNOT chain == Gray-code permutation, ping-pong buffers -------------
    const int nxt = cur ^ 1;
    for (int t = 0; t < 128; ++t) {
      const int j = t * 32 + lane;
      lds[nxt][w][j] = lds[cur][w][j ^ (j >> 1)];
    }
    cur = nxt;
  }

  // ---- p = probs[:1024] / max(probs[:1024])  (sum normalization cancels) --
  float m = 0.0f;
  for (int t = 0; t < 32; ++t) {
    const float sv = lds[cur][w][t * 32 + lane];
    m = fmaxf(m, sv * sv);
  }
#pragma unroll
  for (int off = 16; off > 0; off >>= 1)
    m = fmaxf(m, __shfl_xor(m, off, 32));
  const float inv = 1.0f / m;

  for (int t = 0; t < 32; ++t) {
    const int j = t * 32 + lane;
    const float sv = lds[cur][w][j];
    out[(size_t)b * (NGEN * KOUT) + (size_t)g * KOUT + j] = sv * sv * inv;
  }
}

// ---------------------------------------------------------------------------
extern "C" void kernel_launch(void* const* d_in, const int* in_sizes, int n_in,
                              void* d_out, int out_size, void* d_ws,
                              size_t ws_size, hipStream_t stream) {
  const float* x  = (const float*)d_in[0];   // (B, 12) float32
  const float* qp = (const float*)d_in[1];   // (8, 72) float32
  float* Umats = (float*)d_ws;               // 144 * 256 floats = 144 KB
  float* out   = (float*)d_out;

  const int B = in_sizes[0] / NQ;            // 4096

  qgen_precompute_unitaries<<<NGEN * QDEPTH * 3, 256, 0, stream>>>(qp, Umats);

  dim3 grid(B / WPB, NGEN);
  qgen_simulate<<<grid, BLOCK, 0, stream>>>(x, Umats, out);
}